// EdgeConv_49289044689295
// MI455X (gfx1250) — compile-verified
//
#include <hip/hip_runtime.h>

#define N_NODES 50000
#define N_EDGES 800000
#define D 64
#define N_ROWBLK (N_NODES / 16) // 3125 exact

typedef __attribute__((ext_vector_type(2))) float v2f;
typedef __attribute__((ext_vector_type(8))) float v8f;

// Order-preserving float32 -> uint32 map (total order, negatives < positives).
__device__ __forceinline__ unsigned f2key(float f) {
    unsigned u = __float_as_uint(f);
    return u ^ ((unsigned)(((int)u) >> 31) | 0x80000000u);
}
__device__ __forceinline__ float key2f(unsigned k) {
    unsigned u = k ^ ((unsigned)((~(int)k) >> 31) | 0x80000000u);
    return __uint_as_float(u);
}

// ---------------------------------------------------------------------------
// Kernel 1: A = feat @ Wt ; B = feat @ (Wt + Wp) + (bt + bp)
// One wave per 16-row block; 4 col tiles x 2 outputs; K chained 16x (K=4)
// via V_WMMA_F32_16X16X4_F32. Weights pre-swizzled in LDS into the exact
// per-lane B-fragment layout so each fragment is one aligned ds_load_b64.
//   fragment entry [ks][t][lane] = ( W[4*ks + 2*(lane>>4)    ][16*t + (lane&15)],
//                                    W[4*ks + 2*(lane>>4) + 1][16*t + (lane&15)] )
// ---------------------------------------------------------------------------
__global__ __launch_bounds__(256) void edgeconv_gemm(
    const float* __restrict__ feat,
    const float* __restrict__ Wt, const float* __restrict__ bt,
    const float* __restrict__ Wp, const float* __restrict__ bp,
    float* __restrict__ A, float* __restrict__ B) {
    __shared__ v2f lWt[16 * 4 * 32];   // 16 KB, theta-weight fragments
    __shared__ v2f lWs[16 * 4 * 32];   // 16 KB, (theta+phi)-weight fragments

    for (int idx = threadIdx.x; idx < 16 * 4 * 32; idx += 256) {
        const int ln = idx & 31;
        const int t  = (idx >> 5) & 3;
        const int ks = idx >> 7;
        const int kb = ks * 4 + ((ln >> 4) << 1);
        const int n  = t * 16 + (ln & 15);
        const float wt0 = Wt[kb * D + n];
        const float wt1 = Wt[(kb + 1) * D + n];
        v2f ft, fs;
        ft.x = wt0;
        ft.y = wt1;
        fs.x = wt0 + Wp[kb * D + n];
        fs.y = wt1 + Wp[(kb + 1) * D + n];
        lWt[idx] = ft;
        lWs[idx] = fs;
    }
    __syncthreads();

    int rb = blockIdx.x * 8 + (threadIdx.x >> 5);
    if (rb >= N_ROWBLK) return;               // wave-uniform: EXEC stays all-1s

    const int lane  = threadIdx.x & 31;
    const int lrow  = lane & 15;              // M (A) / N (B,C,D) within tile
    const int khalf = (lane >> 4) * 2;        // lanes 16-31 carry K+2,K+3
    const int row0  = rb * 16;

    v8f acc_t[4] = { {}, {}, {}, {} };
    v8f acc_s[4] = { {}, {}, {}, {} };

    const float* featrow = feat + (size_t)(row0 + lrow) * D;
    for (int ks = 0; ks < 16; ++ks) {
        const int kb = ks * 4 + khalf;        // even -> 8B-aligned b64 load
        v2f a = *(const v2f*)(featrow + kb);
        const int fbase = ks * 128 + lane;
#pragma unroll
        for (int t = 0; t < 4; ++t) {
            v2f bfr = lWt[fbase + t * 32];
            v2f bsr = lWs[fbase + t * 32];
            acc_t[t] = __builtin_amdgcn_wmma_f32_16x16x4_f32(
                false, a, false, bfr, (short)0, acc_t[t], false, false);
            acc_s[t] = __builtin_amdgcn_wmma_f32_16x16x4_f32(
                false, a, false, bsr, (short)0, acc_s[t], false, false);
        }
    }

    const int rbase = row0 + (lane >> 4) * 8; // C/D: VGPR r -> rows r / r+8
#pragma unroll
    for (int t = 0; t < 4; ++t) {
        const int col = t * 16 + lrow;
        const float bias = bt[col] + bp[col];
#pragma unroll
        for (int r = 0; r < 8; ++r) {
            const size_t idx = (size_t)(rbase + r) * D + col;
            A[idx] = acc_t[t][r];
            B[idx] = acc_s[t][r] + bias;
        }
    }
}

// ---------------------------------------------------------------------------
// Kernel 0: init min-key buffer (lives in d_out) to key(+inf); re-run every
// launch so graph replay is deterministic.
// ---------------------------------------------------------------------------
__global__ void edgeconv_init(unsigned* __restrict__ minkey) {
    int i = blockIdx.x * blockDim.x + threadIdx.x;
    if (i < N_NODES * D) minkey[i] = 0xFF800000u; // f2key(+inf)
}

// ---------------------------------------------------------------------------
// Kernel 2: per edge, scatter-min of A[src] into minkey[dst] (L2-resident).
// One thread handles a float4 channel quad: 16 threads per edge.
// ---------------------------------------------------------------------------
__global__ __launch_bounds__(256) void edgeconv_scatter(
    const int* __restrict__ src, const int* __restrict__ dst,
    const float* __restrict__ A, unsigned* __restrict__ minkey) {
    int t = blockIdx.x * 256 + threadIdx.x;
    int e = t >> 4;
    if (e >= N_EDGES) return;
    int q = (t & 15) << 2;
    int s = src[e];
    int d = dst[e];
    const float4 va = *(const float4*)(A + (size_t)s * D + q);
    unsigned* mk = minkey + (size_t)d * D + q;
    atomicMin(mk + 0, f2key(va.x));
    atomicMin(mk + 1, f2key(va.y));
    atomicMin(mk + 2, f2key(va.z));
    atomicMin(mk + 3, f2key(va.w));
}

// ---------------------------------------------------------------------------
// Kernel 3: out = B - unmap(minkey), in place over d_out.
// ---------------------------------------------------------------------------
__global__ void edgeconv_final(const unsigned* __restrict__ minkey,
                               const float* __restrict__ B,
                               float* __restrict__ out) {
    int i = blockIdx.x * blockDim.x + threadIdx.x;
    if (i < N_NODES * D) out[i] = B[i] - key2f(minkey[i]);
}

extern "C" void kernel_launch(void* const* d_in, const int* in_sizes, int n_in,
                              void* d_out, int out_size, void* d_ws, size_t ws_size,
                              hipStream_t stream) {
    const float* feat = (const float*)d_in[0];
    const int*   src  = (const int*)d_in[1];
    const int*   dst  = (const int*)d_in[2];
    const float* Wt   = (const float*)d_in[3];
    const float* bt   = (const float*)d_in[4];
    const float* Wp   = (const float*)d_in[5];
    const float* bp   = (const float*)d_in[6];

    float* A = (float*)d_ws;                       // N*D f32  (12.8 MB)
    float* B = A + (size_t)N_NODES * D;            // N*D f32  (12.8 MB)
    unsigned* minkey = (unsigned*)d_out;           // keyed min lives in d_out

    const int nOut = N_NODES * D;
    edgeconv_init<<<(nOut + 255) / 256, 256, 0, stream>>>(minkey);
    edgeconv_gemm<<<(N_ROWBLK + 7) / 8, 256, 0, stream>>>(feat, Wt, bt, Wp, bp, A, B);
    edgeconv_scatter<<<(N_EDGES * 16) / 256, 256, 0, stream>>>(src, dst, A, minkey);
    edgeconv_final<<<(nOut + 255) / 256, 256, 0, stream>>>(minkey, B, (float*)d_out);
}